// Self_Fusion_1047972020963
// MI455X (gfx1250) — compile-verified
//
#include <hip/hip_runtime.h>
#include <math.h>

typedef __attribute__((ext_vector_type(2))) float v2f;
typedef __attribute__((ext_vector_type(8))) float v8f;

#define BB 32
#define TT 1024
#define DD 256
#define HH 256

// ---------------------------------------------------------------------------
// Stage 1: c = mean(x over valid len); W = MLP_w(c); b = MLP_b(c)
// One 256-thread block per batch. Tiny GEMMs -> LDS dot products.
// ---------------------------------------------------------------------------
__global__ __launch_bounds__(256) void stage1_cond_mlp(
    const float* __restrict__ x, const int* __restrict__ len_x,
    const float* __restrict__ Ww1_w, const float* __restrict__ Ww1_b,
    const float* __restrict__ Ww2_w, const float* __restrict__ Ww2_b,
    const float* __restrict__ Wb1_w, const float* __restrict__ Wb1_b,
    const float* __restrict__ Wb2_w, const float* __restrict__ Wb2_b,
    float* __restrict__ Wvec, float* __restrict__ bvec)
{
    __shared__ float c[DD];
    __shared__ float h[HH];
    const int b = blockIdx.x;
    const int t = threadIdx.x;
    const float L = (float)len_x[b];
    const float* xb = x + (size_t)b * TT * DD;

    float s = 0.0f;
    for (int n = 0; n < TT; ++n) s += xb[n * DD + t];
    c[t] = s / L;
    __syncthreads();

    // W-path layer 1 (exact GELU)
    {
        float acc = Ww1_b[t];
        const float* wr = Ww1_w + t * DD;
        for (int k = 0; k < DD; ++k) acc += wr[k] * c[k];
        h[t] = 0.5f * acc * (1.0f + erff(acc * 0.70710678118654752f));
    }
    __syncthreads();
    // W-path layer 2
    {
        float acc = Ww2_b[t];
        const float* wr = Ww2_w + t * HH;
        for (int k = 0; k < HH; ++k) acc += wr[k] * h[k];
        Wvec[b * DD + t] = acc;
    }
    __syncthreads();
    // b-path layer 1
    {
        float acc = Wb1_b[t];
        const float* wr = Wb1_w + t * DD;
        for (int k = 0; k < DD; ++k) acc += wr[k] * c[k];
        float g = 0.5f * acc * (1.0f + erff(acc * 0.70710678118654752f));
        __syncthreads();
        h[t] = g;
    }
    __syncthreads();
    // b-path layer 2
    {
        float acc = Wb2_b[t];
        const float* wr = Wb2_w + t * HH;
        for (int k = 0; k < HH; ++k) acc += wr[k] * h[k];
        bvec[b * DD + t] = acc;
    }
}

// ---------------------------------------------------------------------------
// Stage 2: forward ragged DFT (real input) + filter + cRelu -> act_{re,im}
// Grid (B, T/16). 4 waves/block; wave w owns columns [64w, 64w+64).
// Hot K-loop is branch-free: full 4-wide steps with unconditional loads
// (always in-bounds for n < Lmain); one masked tail step (loads still
// in-bounds since Lmain+3 <= T-1; A-tile zeros kill invalid products).
// Twiddles via hardware-trig in revolutions: rev = ((k*n) mod L) / L,
// with the mod advanced incrementally (no per-step integer division).
// ---------------------------------------------------------------------------
__global__ __launch_bounds__(128) void fwd_dft_filter(
    const float* __restrict__ x, const int* __restrict__ len_x,
    const float* __restrict__ Wvec, const float* __restrict__ bvec,
    float* __restrict__ act_re, float* __restrict__ act_im)
{
    const int b    = blockIdx.x;
    const int k0   = blockIdx.y * 16;
    const int wave = threadIdx.x >> 5;
    const int lane = threadIdx.x & 31;
    const int half = lane >> 4;
    const int m16  = lane & 15;
    const int L    = len_x[b];
    const float invL = 1.0f / (float)L;
    const float* xb = x + (size_t)b * TT * DD;
    const int d0  = wave * 64;
    const int kk0 = half * 2;           // A/B element j maps to K = kk0 + j
    const int kA  = k0 + m16;           // A-matrix row for this lane

    const v8f vzero = {0.f, 0.f, 0.f, 0.f, 0.f, 0.f, 0.f, 0.f};
    v8f acc_re[4], acc_im[4];
#pragma unroll
    for (int t = 0; t < 4; ++t) { acc_re[t] = vzero; acc_im[t] = vzero; }

    // exact incremental phase index: im_j = (kA * (n + kk0 + j)) mod L
    const int stepm = (kA * 4) % L;
    int im0 = (kA * (kk0 + 0)) % L;
    int im1 = (kA * (kk0 + 1)) % L;

    const int Lmain = L & ~3;           // all nn < L inside [0, Lmain)

    for (int n = 0; n < Lmain; n += 4) {
        v2f a_cos, a_nsin;
        {
            const float r0 = (float)im0 * invL;   // revolutions in [0,1)
            const float r1 = (float)im1 * invL;
            a_cos[0]  =  __builtin_amdgcn_cosf(r0);
            a_cos[1]  =  __builtin_amdgcn_cosf(r1);
            a_nsin[0] = -__builtin_amdgcn_sinf(r0);   // e^{-i theta}
            a_nsin[1] = -__builtin_amdgcn_sinf(r1);
            im0 += stepm; im0 = (im0 >= L) ? im0 - L : im0;
            im1 += stepm; im1 = (im1 >= L) ? im1 - L : im1;
        }
        const int n0 = n + kk0, n1 = n0 + 1;
#pragma unroll
        for (int t = 0; t < 4; ++t) {
            const int d = d0 + t * 16 + m16;
            v2f bx;
            bx[0] = xb[n0 * DD + d];
            bx[1] = xb[n1 * DD + d];
            acc_re[t] = __builtin_amdgcn_wmma_f32_16x16x4_f32(
                false, a_cos,  false, bx, (short)0, acc_re[t], false, false);
            acc_im[t] = __builtin_amdgcn_wmma_f32_16x16x4_f32(
                false, a_nsin, false, bx, (short)0, acc_im[t], false, false);
        }
    }

    if (Lmain < L) {                    // tail: L%4 in {1,2,3}; nn <= 1023
        const int n0 = Lmain + kk0, n1 = n0 + 1;
        v2f a_cos, a_nsin;
        const float r0 = (float)im0 * invL;
        const float r1 = (float)im1 * invL;
        const float c0 =  __builtin_amdgcn_cosf(r0);
        const float c1 =  __builtin_amdgcn_cosf(r1);
        const float s0 = -__builtin_amdgcn_sinf(r0);
        const float s1 = -__builtin_amdgcn_sinf(r1);
        a_cos[0]  = (n0 < L) ? c0 : 0.f;
        a_cos[1]  = (n1 < L) ? c1 : 0.f;
        a_nsin[0] = (n0 < L) ? s0 : 0.f;
        a_nsin[1] = (n1 < L) ? s1 : 0.f;
#pragma unroll
        for (int t = 0; t < 4; ++t) {
            const int d = d0 + t * 16 + m16;
            v2f bx;                      // in-bounds; zero-A masks invalid rows
            bx[0] = xb[n0 * DD + d];
            bx[1] = xb[n1 * DD + d];
            acc_re[t] = __builtin_amdgcn_wmma_f32_16x16x4_f32(
                false, a_cos,  false, bx, (short)0, acc_re[t], false, false);
            acc_im[t] = __builtin_amdgcn_wmma_f32_16x16x4_f32(
                false, a_nsin, false, bx, (short)0, acc_im[t], false, false);
        }
    }

    // filter + cRelu + store (rows k >= L masked to zero)
#pragma unroll
    for (int t = 0; t < 4; ++t) {
        const int d = d0 + t * 16 + m16;
        const float wp = 1.0f + Wvec[b * DD + d];
        const float ba = bvec[b * DD + d];
#pragma unroll
        for (int r = 0; r < 8; ++r) {
            const int k = k0 + r + 8 * half;
            float fr = 0.f, fi = 0.f;
            if (k < L) {
                fr = acc_re[t][r] * wp + ba;
                fi = acc_im[t][r] * wp;
                const float mag2 = fr * fr + fi * fi;
                fr = (mag2 > 0.f) ? fr : 0.f;   // cRelu: mag*e^{i*angle(z)} == z
                fi = (mag2 > 0.f) ? fi : 0.f;
            }
            const size_t idx = (size_t)b * TT * DD + (size_t)k * DD + d;
            act_re[idx] = fr;
            act_im[idx] = fi;
        }
    }
}

// ---------------------------------------------------------------------------
// Stage 3: inverse ragged DFT, real part only, scaled by 1/L.
// out[k,d] = (1/L) * sum_{n<L} ( cos(th)*ar - sin(th)*ai ), th = +2*pi*k*n/L
// Two chained WMMAs into a single accumulator (negated-sin A tile).
// ---------------------------------------------------------------------------
__global__ __launch_bounds__(128) void inv_dft_real(
    const float* __restrict__ act_re, const float* __restrict__ act_im,
    const int* __restrict__ len_x, float* __restrict__ out)
{
    const int b    = blockIdx.x;
    const int k0   = blockIdx.y * 16;
    const int wave = threadIdx.x >> 5;
    const int lane = threadIdx.x & 31;
    const int half = lane >> 4;
    const int m16  = lane & 15;
    const int L    = len_x[b];
    const float invL = 1.0f / (float)L;
    const size_t base = (size_t)b * TT * DD;
    const int d0  = wave * 64;
    const int kk0 = half * 2;
    const int kA  = k0 + m16;

    const v8f vzero = {0.f, 0.f, 0.f, 0.f, 0.f, 0.f, 0.f, 0.f};
    v8f acc[4];
#pragma unroll
    for (int t = 0; t < 4; ++t) acc[t] = vzero;

    const int stepm = (kA * 4) % L;
    int im0 = (kA * (kk0 + 0)) % L;
    int im1 = (kA * (kk0 + 1)) % L;

    const int Lmain = L & ~3;

    for (int n = 0; n < Lmain; n += 4) {
        v2f a_cos, a_nsin;
        {
            const float r0 = (float)im0 * invL;
            const float r1 = (float)im1 * invL;
            a_cos[0]  =  __builtin_amdgcn_cosf(r0);
            a_cos[1]  =  __builtin_amdgcn_cosf(r1);
            a_nsin[0] = -__builtin_amdgcn_sinf(r0);
            a_nsin[1] = -__builtin_amdgcn_sinf(r1);
            im0 += stepm; im0 = (im0 >= L) ? im0 - L : im0;
            im1 += stepm; im1 = (im1 >= L) ? im1 - L : im1;
        }
        const int n0 = n + kk0, n1 = n0 + 1;
#pragma unroll
        for (int t = 0; t < 4; ++t) {
            const int d = d0 + t * 16 + m16;
            v2f bar, bai;
            bar[0] = act_re[base + (size_t)(n0 * DD + d)];
            bar[1] = act_re[base + (size_t)(n1 * DD + d)];
            bai[0] = act_im[base + (size_t)(n0 * DD + d)];
            bai[1] = act_im[base + (size_t)(n1 * DD + d)];
            acc[t] = __builtin_amdgcn_wmma_f32_16x16x4_f32(
                false, a_cos,  false, bar, (short)0, acc[t], false, false);
            acc[t] = __builtin_amdgcn_wmma_f32_16x16x4_f32(
                false, a_nsin, false, bai, (short)0, acc[t], false, false);
        }
    }

    if (Lmain < L) {
        const int n0 = Lmain + kk0, n1 = n0 + 1;
        v2f a_cos, a_nsin;
        const float r0 = (float)im0 * invL;
        const float r1 = (float)im1 * invL;
        const float c0 =  __builtin_amdgcn_cosf(r0);
        const float c1 =  __builtin_amdgcn_cosf(r1);
        const float s0 = -__builtin_amdgcn_sinf(r0);
        const float s1 = -__builtin_amdgcn_sinf(r1);
        a_cos[0]  = (n0 < L) ? c0 : 0.f;
        a_cos[1]  = (n1 < L) ? c1 : 0.f;
        a_nsin[0] = (n0 < L) ? s0 : 0.f;
        a_nsin[1] = (n1 < L) ? s1 : 0.f;
#pragma unroll
        for (int t = 0; t < 4; ++t) {
            const int d = d0 + t * 16 + m16;
            v2f bar, bai;
            bar[0] = act_re[base + (size_t)(n0 * DD + d)];
            bar[1] = act_re[base + (size_t)(n1 * DD + d)];
            bai[0] = act_im[base + (size_t)(n0 * DD + d)];
            bai[1] = act_im[base + (size_t)(n1 * DD + d)];
            acc[t] = __builtin_amdgcn_wmma_f32_16x16x4_f32(
                false, a_cos,  false, bar, (short)0, acc[t], false, false);
            acc[t] = __builtin_amdgcn_wmma_f32_16x16x4_f32(
                false, a_nsin, false, bai, (short)0, acc[t], false, false);
        }
    }

#pragma unroll
    for (int t = 0; t < 4; ++t) {
        const int d = d0 + t * 16 + m16;
#pragma unroll
        for (int r = 0; r < 8; ++r) {
            const int k = k0 + r + 8 * half;
            const float v = (k < L) ? acc[t][r] * invL : 0.f;
            out[base + (size_t)k * DD + d] = v;
        }
    }
}

// ---------------------------------------------------------------------------
extern "C" void kernel_launch(void* const* d_in, const int* in_sizes, int n_in,
                              void* d_out, int out_size, void* d_ws, size_t ws_size,
                              hipStream_t stream) {
    const float* x     = (const float*)d_in[0];
    const int*   len_x = (const int*)  d_in[1];
    const float* Ww1_w = (const float*)d_in[2];
    const float* Ww1_b = (const float*)d_in[3];
    const float* Ww2_w = (const float*)d_in[4];
    const float* Ww2_b = (const float*)d_in[5];
    const float* Wb1_w = (const float*)d_in[6];
    const float* Wb1_b = (const float*)d_in[7];
    const float* Wb2_w = (const float*)d_in[8];
    const float* Wb2_b = (const float*)d_in[9];

    float* ws     = (float*)d_ws;
    float* Wvec   = ws;                                   // B*D
    float* bvec   = Wvec + BB * DD;                       // B*D
    float* act_re = bvec + BB * DD;                       // B*T*D
    float* act_im = act_re + (size_t)BB * TT * DD;        // B*T*D

    stage1_cond_mlp<<<BB, 256, 0, stream>>>(
        x, len_x, Ww1_w, Ww1_b, Ww2_w, Ww2_b, Wb1_w, Wb1_b, Wb2_w, Wb2_b,
        Wvec, bvec);

    dim3 grid(BB, TT / 16);
    fwd_dft_filter<<<grid, 128, 0, stream>>>(x, len_x, Wvec, bvec, act_re, act_im);
    inv_dft_real<<<grid, 128, 0, stream>>>(act_re, act_im, len_x, (float*)d_out);
}